// DynamicFilter_24927990186532
// MI455X (gfx1250) — compile-verified
//
#include <hip/hip_runtime.h>

// CDNA5 / gfx1250: fused dynamic-filter (1x1-conv GEMM + 3x3 weighted gather)
// GEMM on the fp32 matrix cores: v_wmma_f32_16x16x4_f32.
// v2: 2 N-subtiles per wave (A-fragment LDS loads amortized over 2 WMMAs),
//     A-fragments loaded as aligned 8B vectors -> ds_load_b64 with imm offsets.

typedef __attribute__((ext_vector_type(2))) float v2f;
typedef __attribute__((ext_vector_type(8))) float v8f;

namespace {
constexpr int kBimg = 4;
constexpr int kC    = 256;
constexpr int kH    = 128;
constexpr int kW    = 128;
constexpr int kHW   = kH * kW;              // 16384
constexpr long kCHW = (long)kC * kHW;       // 4194304
constexpr int kKK   = 9;                    // 3x3 positions
constexpr int kKD   = kC;                   // GEMM reduction dim = 256
constexpr int TILE_C = 16;                  // channels per workgroup -> 144 GEMM rows
constexpr int NSUB   = 2;                   // N-subtiles (16 px) per wave
constexpr int TILE_N = 128;                 // pixels per workgroup (32 per wave)
constexpr int ROWS   = TILE_C * kKK;        // 144
constexpr int LDSROW = kKD + 2;             // padded LDS row stride (floats), conflict-free
}

__launch_bounds__(128)
__global__ void dynfilter_wmma(const float* __restrict__ x,
                               const float* __restrict__ Wm,
                               const float* __restrict__ bias,
                               float* __restrict__ out)
{
    extern __shared__ float ldsW[];         // ROWS * LDSROW floats = 148608 B

    const int tid  = threadIdx.x;
    const int lane = tid & 31;
    const int wave = tid >> 5;
    const int l16  = lane & 15;
    const int hi   = lane >> 4;             // 0: lanes 0-15, 1: lanes 16-31

    const int c0 = blockIdx.y * TILE_C;     // first channel of this WG
    const int n0 = blockIdx.x * TILE_N;     // first pixel of this WG

    // ---- stage the contiguous W tile rows [c0*9, c0*9+144) x K=256 into LDS ----
    {
        const float* src = Wm + (long)c0 * kKK * kKD;
        for (int i = tid; i < ROWS * kKD; i += 128) {
            int r = i >> 8;                 // / 256
            int k = i & 255;
            ldsW[r * LDSROW + k] = src[i];
        }
    }
    __syncthreads();

    // ---- this lane's two pixels (columns of B, columns of output) ----
    // TILE_N = 128 = one image row => same b, h for the whole workgroup tile.
    const int nA   = n0 + wave * (16 * NSUB) + l16;   // subtile 0
    const int nBp  = nA + 16;                          // subtile 1
    const int bImg = nA / kHW;
    const int hwA  = nA % kHW;
    const int h    = hwA / kW;
    const int wA   = hwA % kW;
    const float* xcol0 = x + (long)bImg * kCHW + hwA;        // + k*kHW per k
    const float* xcol1 = xcol0 + 16;                         // same row, w+16

    // ---- init accumulators with bias (F = W*X + b) ----
    // C/D layout: VGPR v, lanes 0-15 -> M=v ; lanes 16-31 -> M=v+8 ; N = lane%16
    v8f acc[NSUB][kKK];
    #pragma unroll
    for (int p = 0; p < kKK; ++p) {
        #pragma unroll
        for (int v = 0; v < 8; ++v) {
            const int c = c0 + v + 8 * hi;
            const float bv = bias[c * kKK + p];
            acc[0][p][v] = bv;
            acc[1][p][v] = bv;
        }
    }

    // A layout (16x4 f32): lane m=l16 holds row; VGPR0 = K0 (lo lanes)/K2 (hi lanes),
    // VGPR1 = K1/K3 -> consecutive pair at k0 + 2*hi.
    // M-tile p maps lane l16 -> GEMM row (c0+l16)*9 + p == LDS row l16*9 + p.
    // Base offset is even-dword => 8B-aligned; per-p/k0 deltas fit ds b64 imm offsets.
    const float* ldsBase = &ldsW[(l16 * kKK) * LDSROW + 2 * hi];

    for (int k0 = 0; k0 < kKD; k0 += 4) {
        // B fragments (4x16 f32): lanes 0-15 carry K=k0,k0+1 ; lanes 16-31 K=k0+2,k0+3
        const int kb = k0 + 2 * hi;
        v2f b0, b1;
        b0.x = xcol0[(long)kb * kHW];
        b0.y = xcol0[(long)(kb + 1) * kHW];
        b1.x = xcol1[(long)kb * kHW];
        b1.y = xcol1[(long)(kb + 1) * kHW];
        #pragma unroll
        for (int p = 0; p < kKK; ++p) {
            const v2f a = *reinterpret_cast<const v2f*>(ldsBase + p * LDSROW + k0);
            // 8 args: (neg_a, A, neg_b, B, c_mod, C, reuse_a, reuse_b)
            acc[0][p] = __builtin_amdgcn_wmma_f32_16x16x4_f32(
                false, a, false, b0, (short)0, acc[0][p], false, false);
            acc[1][p] = __builtin_amdgcn_wmma_f32_16x16x4_f32(
                false, a, false, b1, (short)0, acc[1][p], false, false);
        }
    }

    // ---- fused 3x3 dynamic-filter epilogue ----
    // patch p = i*3 + j (row-major), value = x[h+i-1, w+j-1] with zero padding.
    #pragma unroll
    for (int t = 0; t < NSUB; ++t) {
        const int w  = wA + t * 16;
        const int hw = hwA + t * 16;
        #pragma unroll
        for (int v = 0; v < 8; ++v) {
            const int c = c0 + v + 8 * hi;
            const float* xc = x + (long)bImg * kCHW + (long)c * kHW;
            float sum = 0.0f;
            #pragma unroll
            for (int i = 0; i < 3; ++i) {
                const int hh = h + i - 1;
                #pragma unroll
                for (int j = 0; j < 3; ++j) {
                    const int ww = w + j - 1;
                    float xv = 0.0f;
                    if (hh >= 0 && hh < kH && ww >= 0 && ww < kW)
                        xv = xc[hh * kW + ww];
                    sum += acc[t][i * 3 + j][v] * xv;
                }
            }
            out[(long)bImg * kCHW + (long)c * kHW + hw] = sum;
        }
    }
}

extern "C" void kernel_launch(void* const* d_in, const int* in_sizes, int n_in,
                              void* d_out, int out_size, void* d_ws, size_t ws_size,
                              hipStream_t stream) {
    const float* x    = (const float*)d_in[0];   // (4,256,128,128)
    const float* Wm   = (const float*)d_in[1];   // (2304,256)
    const float* bias = (const float*)d_in[2];   // (2304,)
    float* out = (float*)d_out;                  // (4,256,128,128)

    dim3 grid((kBimg * kHW) / TILE_N, kC / TILE_C);   // (512, 16)
    dim3 block(128);
    size_t shmem = (size_t)ROWS * LDSROW * sizeof(float);  // 148608 B
    dynfilter_wmma<<<grid, block, shmem, stream>>>(x, Wm, bias, out);
}